// SlotGenerator_32418413150613
// MI455X (gfx1250) — compile-verified
//
#include <hip/hip_runtime.h>
#include <hip/hip_bf16.h>

typedef __attribute__((ext_vector_type(2))) float v2f;
typedef __attribute__((ext_vector_type(8))) float v8f;

#define TPB 256

// ---------------------------------------------------------------------------
// Init: w[row] = sum_l E[slot[j][l]]  (row = b*J + j),  h[row] = hidden[b]
// ---------------------------------------------------------------------------
__global__ void init_wh(const float* __restrict__ E, const int* __restrict__ slot,
                        const float* __restrict__ hidden,
                        float* __restrict__ w, float* __restrict__ h,
                        int J, int Ls, int H, int BJ) {
  int row = blockIdx.x;
  if (row >= BJ) return;
  int b = row / J, j = row % J;
  for (int c = threadIdx.x; c < H; c += blockDim.x) {
    float acc = 0.f;
    for (int l = 0; l < Ls; ++l)
      acc += E[(long)slot[j * Ls + l] * H + c];
    w[(long)row * H + c] = acc;
    h[(long)row * H + c] = hidden[(long)b * H + c];
  }
}

// ---------------------------------------------------------------------------
// Generic NT-GEMM on fp32 WMMA:  out[M,N] = X[M,K] @ W[N,K]^T
// Used for: vocab logits (W = E), GRU gates (W = W_ih / W_hh).
// One wave per 32(M) x 64(N) tile: 2 M-tiles x 4 N-tiles = 8 accumulators.
// K is chunked by 16: 24 b64 loads batched in flight, then 32 WMMAs, so load
// latency is paid once per 32 matrix ops instead of once per op.
// A layout: lane 0-15 -> M=lane, VGPR0/1 = K0/K1; lanes 16-31 -> K2/K3.
// B layout mirrors A over N.  C/D: VGPR r -> M=r (low half) / M=8+r (high).
// ---------------------------------------------------------------------------
__global__ void gemm_nt_wmma(const float* __restrict__ X, const float* __restrict__ W,
                             float* __restrict__ out, int K, int N, int M, int MT2) {
  int task = blockIdx.x;
  int mt = task % MT2, nt = task / MT2;   // mt fastest: consecutive waves share B
  int m0 = mt * 32, n0 = nt * 64;
  int lane = threadIdx.x & 31;
  int half = lane >> 4, l16 = lane & 15;

  int mA = m0 + l16;      if (mA >= M) mA = M - 1;   // clamp loads; stores guarded
  int mB = m0 + 16 + l16; if (mB >= M) mB = M - 1;
  const float* xr0 = X + (long)mA * K + 2 * half;
  const float* xr1 = X + (long)mB * K + 2 * half;
  const float* wr0;
  const float* wr1;
  const float* wr2;
  const float* wr3;
  {
    int n;
    n = n0 + 0 * 16 + l16; if (n >= N) n = N - 1; wr0 = W + (long)n * K + 2 * half;
    n = n0 + 1 * 16 + l16; if (n >= N) n = N - 1; wr1 = W + (long)n * K + 2 * half;
    n = n0 + 2 * 16 + l16; if (n >= N) n = N - 1; wr2 = W + (long)n * K + 2 * half;
    n = n0 + 3 * 16 + l16; if (n >= N) n = N - 1; wr3 = W + (long)n * K + 2 * half;
  }

  v8f acc[2][4];
#pragma unroll
  for (int i = 0; i < 2; ++i)
#pragma unroll
    for (int j = 0; j < 4; ++j) acc[i][j] = (v8f){};

  for (int k = 0; k < K; k += 16) {
    v2f a0[4], a1[4], bb[4][4];
#pragma unroll
    for (int kk = 0; kk < 4; ++kk) {
      a0[kk] = *(const v2f*)(xr0 + k + 4 * kk);
      a1[kk] = *(const v2f*)(xr1 + k + 4 * kk);
    }
#pragma unroll
    for (int kk = 0; kk < 4; ++kk) {
      bb[0][kk] = *(const v2f*)(wr0 + k + 4 * kk);
      bb[1][kk] = *(const v2f*)(wr1 + k + 4 * kk);
      bb[2][kk] = *(const v2f*)(wr2 + k + 4 * kk);
      bb[3][kk] = *(const v2f*)(wr3 + k + 4 * kk);
    }
#pragma unroll
    for (int kk = 0; kk < 4; ++kk) {
#pragma unroll
      for (int j = 0; j < 4; ++j) {
        acc[0][j] = __builtin_amdgcn_wmma_f32_16x16x4_f32(
            false, a0[kk], false, bb[j][kk], (short)0, acc[0][j], false, false);
        acc[1][j] = __builtin_amdgcn_wmma_f32_16x16x4_f32(
            false, a1[kk], false, bb[j][kk], (short)0, acc[1][j], false, false);
      }
    }
  }

#pragma unroll
  for (int i = 0; i < 2; ++i) {
#pragma unroll
    for (int r = 0; r < 8; ++r) {
      int m = m0 + 16 * i + r + 8 * half;
      if (m >= M) continue;
      long rb = (long)m * N;
#pragma unroll
      for (int j = 0; j < 4; ++j) {
        int n = n0 + 16 * j + l16;
        if (n < N) out[rb + n] = acc[i][j][r];
      }
    }
  }
}

// h <- (1-z)*n + z*h  with biases folded in here (PyTorch GRUCell math)
__global__ void gru_combine(const float* __restrict__ gi, const float* __restrict__ gh,
                            const float* __restrict__ b_ih, const float* __restrict__ b_hh,
                            float* __restrict__ h, int H, long BJ) {
  long i = blockIdx.x * (long)blockDim.x + threadIdx.x;
  long tot = BJ * H;
  if (i >= tot) return;
  long row = i / H;
  int c = (int)(i % H);
  long base = row * 3 * (long)H;
  float ir = gi[base + c] + b_ih[c];
  float iz = gi[base + H + c] + b_ih[H + c];
  float in = gi[base + 2 * H + c] + b_ih[2 * H + c];
  float hr = gh[base + c] + b_hh[c];
  float hz = gh[base + H + c] + b_hh[H + c];
  float hn = gh[base + 2 * H + c] + b_hh[2 * H + c];
  float r = 1.f / (1.f + expf(-(ir + hr)));
  float z = 1.f / (1.f + expf(-(iz + hz)));
  float n = tanhf(in + r * hn);
  float ho = h[i];
  h[i] = (1.f - z) * n + z * ho;
}

// ---------------------------------------------------------------------------
// Attention: scores over S, masked softmax, context = attn @ enc  (per row)
// ---------------------------------------------------------------------------
__global__ void attention_kernel(const float* __restrict__ enc, const float* __restrict__ h,
                                 const int* __restrict__ masks,
                                 float* __restrict__ attn, float* __restrict__ ctx,
                                 int S, int H, int J) {
  int row = blockIdx.x;
  int b = row / J;
  extern __shared__ float lds[];           // [H + S + TPB]
  float* hl = lds;
  float* sc = lds + H;
  float* red = lds + H + S;
  const float* eb = enc + (long)b * S * H;
  for (int c = threadIdx.x; c < H; c += blockDim.x) hl[c] = h[(long)row * H + c];
  __syncthreads();
  for (int s = threadIdx.x; s < S; s += blockDim.x) {
    const float* es = eb + (long)s * H;
    float d = 0.f;
    for (int k = 0; k < H; ++k) d += es[k] * hl[k];
    if (masks[(long)b * S + s] != 1) d = -1e9f;
    sc[s] = d;
  }
  __syncthreads();
  float m = -INFINITY;
  for (int s = threadIdx.x; s < S; s += blockDim.x) m = fmaxf(m, sc[s]);
  red[threadIdx.x] = m;
  __syncthreads();
  for (int off = blockDim.x / 2; off > 0; off >>= 1) {
    if (threadIdx.x < off) red[threadIdx.x] = fmaxf(red[threadIdx.x], red[threadIdx.x + off]);
    __syncthreads();
  }
  m = red[0];
  __syncthreads();
  float sum = 0.f;
  for (int s = threadIdx.x; s < S; s += blockDim.x) {
    float e = expf(sc[s] - m);
    sc[s] = e;
    sum += e;
  }
  red[threadIdx.x] = sum;
  __syncthreads();
  for (int off = blockDim.x / 2; off > 0; off >>= 1) {
    if (threadIdx.x < off) red[threadIdx.x] += red[threadIdx.x + off];
    __syncthreads();
  }
  float inv = 1.f / red[0];
  __syncthreads();
  for (int s = threadIdx.x; s < S; s += blockDim.x) {
    sc[s] *= inv;
    attn[(long)row * S + s] = sc[s];
  }
  __syncthreads();
  for (int c = threadIdx.x; c < H; c += blockDim.x) {
    float acc = 0.f;
    for (int s = 0; s < S; ++s) acc += sc[s] * eb[(long)s * H + c];
    ctx[(long)row * H + c] = acc;
  }
}

// ---------------------------------------------------------------------------
// Per-row softmax statistics over V (max, sum of exp)
// ---------------------------------------------------------------------------
__global__ void row_stats(const float* __restrict__ logits,
                          float* __restrict__ rowmax, float* __restrict__ rowsum, int V) {
  int row = blockIdx.x;
  __shared__ float red[TPB];
  const float* lr = logits + (long)row * V;
  float m = -INFINITY;
  for (int v = threadIdx.x; v < V; v += blockDim.x) m = fmaxf(m, lr[v]);
  red[threadIdx.x] = m;
  __syncthreads();
  for (int off = blockDim.x / 2; off > 0; off >>= 1) {
    if (threadIdx.x < off) red[threadIdx.x] = fmaxf(red[threadIdx.x], red[threadIdx.x + off]);
    __syncthreads();
  }
  m = red[0];
  __syncthreads();
  float s = 0.f;
  for (int v = threadIdx.x; v < V; v += blockDim.x) s += expf(lr[v] - m);
  red[threadIdx.x] = s;
  __syncthreads();
  for (int off = blockDim.x / 2; off > 0; off >>= 1) {
    if (threadIdx.x < off) red[threadIdx.x] += red[threadIdx.x + off];
    __syncthreads();
  }
  if (threadIdx.x == 0) {
    rowmax[row] = m;
    rowsum[row] = red[0];
  }
}

// p_gen = sigmoid(concat(w,h,ctx) . W_gen + b_gen)   (W_gen is [3H,1])
__global__ void pgen_kernel(const float* __restrict__ w, const float* __restrict__ h,
                            const float* __restrict__ ctx, const float* __restrict__ W_gen,
                            const float* __restrict__ b_gen, float* __restrict__ pgen, int H) {
  int row = blockIdx.x;
  __shared__ float red[TPB];
  float s = 0.f;
  for (int c = threadIdx.x; c < H; c += blockDim.x) {
    s += w[(long)row * H + c] * W_gen[c]
       + h[(long)row * H + c] * W_gen[H + c]
       + ctx[(long)row * H + c] * W_gen[2 * H + c];
  }
  red[threadIdx.x] = s;
  __syncthreads();
  for (int off = blockDim.x / 2; off > 0; off >>= 1) {
    if (threadIdx.x < off) red[threadIdx.x] += red[threadIdx.x + off];
    __syncthreads();
  }
  if (threadIdx.x == 0) pgen[row] = 1.f / (1.f + expf(-(red[0] + b_gen[0])));
}

// out[row, t, v] = p_gen * softmax(logits)[v]
__global__ void write_probs(const float* __restrict__ logits, const float* __restrict__ rowmax,
                            const float* __restrict__ rowsum, const float* __restrict__ pgen,
                            float* __restrict__ out, int V, int maxlen, int t, long BJ) {
  long i = blockIdx.x * (long)blockDim.x + threadIdx.x;
  long tot = BJ * (long)V;
  if (i >= tot) return;
  long row = i / V;
  int v = (int)(i % V);
  float prob = expf(logits[i] - rowmax[row]) / rowsum[row];
  out[(row * maxlen + t) * (long)V + v] = pgen[row] * prob;
}

// out[row, t, ids[b,s]] += (1-p_gen) * attn[row,s]
__global__ void scatter_ptr(const float* __restrict__ attn, const int* __restrict__ ids,
                            const float* __restrict__ pgen, float* __restrict__ out,
                            int S, int V, int maxlen, int t, int J) {
  int row = blockIdx.x;
  int b = row / J;
  float q = 1.f - pgen[row];
  float* orow = out + ((long)row * maxlen + t) * V;
  for (int s = threadIdx.x; s < S; s += blockDim.x) {
    int id = ids[(long)b * S + s];
    atomicAdd(&orow[id], q * attn[(long)row * S + s]);
  }
}

// argmax over p_final row, then w[row] = E[argmax]  (first-index tie break)
__global__ void argmax_gather(const float* __restrict__ out, const float* __restrict__ E,
                              float* __restrict__ w, int V, int H, int maxlen, int t) {
  int row = blockIdx.x;
  __shared__ float rv[TPB];
  __shared__ int ri[TPB];
  const float* orow = out + ((long)row * maxlen + t) * V;
  float bv = -INFINITY;
  int bi = 0;
  for (int v = threadIdx.x; v < V; v += blockDim.x) {
    float x = orow[v];
    if (x > bv) { bv = x; bi = v; }
  }
  rv[threadIdx.x] = bv;
  ri[threadIdx.x] = bi;
  __syncthreads();
  for (int off = blockDim.x / 2; off > 0; off >>= 1) {
    if (threadIdx.x < off) {
      float ov = rv[threadIdx.x + off];
      int oi = ri[threadIdx.x + off];
      if (ov > rv[threadIdx.x] || (ov == rv[threadIdx.x] && oi < ri[threadIdx.x])) {
        rv[threadIdx.x] = ov;
        ri[threadIdx.x] = oi;
      }
    }
    __syncthreads();
  }
  int idx = ri[0];
  for (int c = threadIdx.x; c < H; c += blockDim.x)
    w[(long)row * H + c] = E[(long)idx * H + c];
}

// gate_logit = context(step 0) @ W_gate + b_gate   (W_gate is [H,G])
__global__ void gate_kernel(const float* __restrict__ ctx, const float* __restrict__ W_gate,
                            const float* __restrict__ b_gate, float* __restrict__ out_g,
                            int H, int G, long BJ) {
  long i = blockIdx.x * (long)blockDim.x + threadIdx.x;
  long tot = BJ * G;
  if (i >= tot) return;
  long row = i / G;
  int g = (int)(i % G);
  const float* cr = ctx + row * H;
  float s = 0.f;
  for (int c = 0; c < H; ++c) s += cr[c] * W_gate[(long)c * G + g];
  out_g[i] = s + b_gate[g];
}

// ---------------------------------------------------------------------------
extern "C" void kernel_launch(void* const* d_in, const int* in_sizes, int n_in,
                              void* d_out, int out_size, void* d_ws, size_t ws_size,
                              hipStream_t stream) {
  (void)n_in; (void)ws_size;
  const int*   ids    = (const int*)  d_in[0];
  const float* enc    = (const float*)d_in[1];
  const float* hidden = (const float*)d_in[2];
  const int*   masks  = (const int*)  d_in[3];
  const int*   slot   = (const int*)  d_in[4];
  const float* E      = (const float*)d_in[6];
  const float* W_ih   = (const float*)d_in[7];
  const float* W_hh   = (const float*)d_in[8];
  const float* b_ih   = (const float*)d_in[9];
  const float* b_hh   = (const float*)d_in[10];
  const float* W_gen  = (const float*)d_in[11];
  const float* b_gen  = (const float*)d_in[12];
  const float* W_gate = (const float*)d_in[13];
  const float* b_gate = (const float*)d_in[14];

  // Derive shapes host-side (max_len lives on-device; recover it from out_size)
  const int H  = in_sizes[9] / 3;          // b_ih = [3H]
  const int G  = in_sizes[14];             // b_gate = [G]
  const int B  = in_sizes[2] / H;          // hidden = [1,B,H]
  const int S  = in_sizes[0] / B;          // input_ids = [B,S]
  const int Ls = 4;                        // slot = [J,4]
  const int J  = in_sizes[4] / Ls;
  const int V  = in_sizes[6] / H;          // E = [V,H]
  const long BJ = (long)B * J;
  const int maxlen = (int)(((long)out_size - BJ * G) / (BJ * (long)V));

  // Carve workspace (256B aligned regions)
  char* p = (char*)d_ws;
  auto carve = [&](size_t bytes) {
    void* r = (void*)p;
    p += (bytes + 255) & ~(size_t)255;
    return r;
  };
  float* w      = (float*)carve(BJ * (size_t)H * 4);
  float* h      = (float*)carve(BJ * (size_t)H * 4);
  float* gi     = (float*)carve(BJ * (size_t)(3 * H) * 4);
  float* gh     = (float*)carve(BJ * (size_t)(3 * H) * 4);
  float* attn   = (float*)carve(BJ * (size_t)S * 4);
  float* ctx    = (float*)carve(BJ * (size_t)H * 4);
  float* logits = (float*)carve(BJ * (size_t)V * 4);
  float* rowmax = (float*)carve(BJ * 4);
  float* rowsum = (float*)carve(BJ * 4);
  float* pg     = (float*)carve(BJ * 4);

  float* out   = (float*)d_out;
  float* out_g = out + BJ * (long)maxlen * V;

  init_wh<<<(int)BJ, TPB, 0, stream>>>(E, slot, hidden, w, h, J, Ls, H, (int)BJ);

  const int H3  = 3 * H;
  const int MT2 = (int)((BJ + 31) / 32);
  const int NTv = (V + 63) / 64;
  const int NTg = (H3 + 63) / 64;
  const long totH = BJ * (long)H;
  const long totV = BJ * (long)V;
  const long totG = BJ * (long)G;

  for (int t = 0; t < maxlen; ++t) {
    // GRU gate GEMMs on the matrix pipe (biases folded into combine)
    gemm_nt_wmma<<<MT2 * NTg, 32, 0, stream>>>(w, W_ih, gi, H, H3, (int)BJ, MT2);
    gemm_nt_wmma<<<MT2 * NTg, 32, 0, stream>>>(h, W_hh, gh, H, H3, (int)BJ, MT2);
    gru_combine<<<(int)((totH + TPB - 1) / TPB), TPB, 0, stream>>>(
        gi, gh, b_ih, b_hh, h, H, BJ);
    attention_kernel<<<(int)BJ, TPB, (H + S + TPB) * sizeof(float), stream>>>(
        enc, h, masks, attn, ctx, S, H, J);
    // Dominant op: vocab projection h @ E^T on fp32 WMMA
    gemm_nt_wmma<<<MT2 * NTv, 32, 0, stream>>>(h, E, logits, H, V, (int)BJ, MT2);
    row_stats<<<(int)BJ, TPB, 0, stream>>>(logits, rowmax, rowsum, V);
    pgen_kernel<<<(int)BJ, TPB, 0, stream>>>(w, h, ctx, W_gen, b_gen, pg, H);
    write_probs<<<(int)((totV + TPB - 1) / TPB), TPB, 0, stream>>>(
        logits, rowmax, rowsum, pg, out, V, maxlen, t, BJ);
    scatter_ptr<<<(int)BJ, TPB, 0, stream>>>(attn, ids, pg, out, S, V, maxlen, t, J);
    argmax_gather<<<(int)BJ, TPB, 0, stream>>>(out, E, w, V, H, maxlen, t);
    if (t == 0) {
      gate_kernel<<<(int)((totG + TPB - 1) / TPB), TPB, 0, stream>>>(
          ctx, W_gate, b_gate, out_g, H, G, BJ);
    }
  }
}